// LRGCN_Batch_68109591380388
// MI455X (gfx1250) — compile-verified
//
#include <hip/hip_runtime.h>
#include <cmath>

typedef __attribute__((ext_vector_type(2))) float v2f;
typedef __attribute__((ext_vector_type(4))) float v4f;
typedef __attribute__((ext_vector_type(8))) float v8f;

constexpr int N_NODES = 50000;
constexpr int KNBR    = 16;
constexpr int ROWS    = KNBR + 1;   // 17 entries per adjacency row
constexpr int NFEAT   = 128;
constexpr int NHID    = 128;
constexpr int NCLASS  = 64;

// XOR swizzle on 4-float granularity: keeps float4 stores and the 2-float
// (8B-aligned) WMMA A-fragment reads contiguous, while spreading the
// stride-128 row accesses across LDS banks.
__device__ __forceinline__ int swz(int row, int col) {
  return col ^ ((row & 15) << 2);
}

// One fused "gather-aggregate -> GEMM(+optional ELU)" stage.
//   dst[n, :] = act( (1/16 * sum_k src[idx[n,k+1]] * wgt[n,k+1]) @ W )
// NODES_PER_BLOCK = 16 * WAVES ; each wave owns a 16-wide column tile of W
// (WAVES*16 == NOUT) and computes it for every 16-node tile of the block.
template <int NODES_PER_BLOCK, int WAVES, int NIN, int NOUT, bool ELU>
__global__ __launch_bounds__(WAVES * 32)
void lrgcn_stage(const float* __restrict__ src,   // gather source [*, NIN]
                 const int*   __restrict__ idx,   // [N, 17]
                 const float* __restrict__ wgt,   // [N, 17]
                 const float* __restrict__ W,     // [NIN, NOUT] row-major
                 float*       __restrict__ dst)   // [N, NOUT]
{
  __shared__ float sAgg[NODES_PER_BLOCK * NIN];   // 64KB (s1) / 32KB (s2)

  const int lane = threadIdx.x & 31;
  const int wave = threadIdx.x >> 5;
  const int lo   = lane & 15;
  const int hi   = lane >> 4;
  const int blockNode = blockIdx.x * NODES_PER_BLOCK;

  // ---- B fragments for this wave's 16 output columns, held in VGPRs.
  // f32 WMMA B layout (4x16): lane l, vgpr v -> B[K = v + 2*(l/16)][N = l%16]
  const int col = (wave << 4) + lo;
  v2f b[NIN / 4];
#pragma unroll
  for (int kk = 0; kk < NIN / 4; ++kk) {
    const int kr = 4 * kk + 2 * hi;
    b[kk].x = W[(kr + 0) * NOUT + col];
    b[kk].y = W[(kr + 1) * NOUT + col];
  }

  // ---- Gather + weighted mean for this wave's 16 nodes into LDS.
  const int aggBase = blockNode + (wave << 4);
  if (aggBase < N_NODES) {                  // wave-uniform guard
    const int fb = lane << 2;               // this lane's 4 features
    for (int m = 0; m < 16; ++m) {
      const int node = aggBase + m;
      const int ib   = node * ROWS + 1;     // skip self entry (idx[:,1:])
      const int   myj = idx[ib + lo];       // lanes 0..15 hold 16 neighbors
      const float myw = wgt[ib + lo];
      v4f acc = {0.f, 0.f, 0.f, 0.f};
#pragma unroll
      for (int k = 0; k < KNBR; ++k) {
        const int   j  = __shfl(myj, k);    // wave32 broadcast
        const float wv = __shfl(myw, k);
        const v4f xv = *(const v4f*)(src + (size_t)j * NIN + fb);
        acc += wv * xv;                     // coalesced 512B row reads
      }
      acc *= (1.0f / 16.0f);                // jnp.mean over K=16
      const int r = (wave << 4) + m;
      *(v4f*)(&sAgg[r * NIN + swz(r, fb)]) = acc;
    }
  }
  __syncthreads();

  // ---- GEMM: 16x16 output tiles via V_WMMA_F32_16X16X4_F32 (full fp32).
#pragma unroll 1
  for (int nt = 0; nt < NODES_PER_BLOCK / 16; ++nt) {
    const int tileNode = blockNode + (nt << 4);
    if (tileNode >= N_NODES) break;         // wave-uniform (N % 16 == 0)
    v8f c = {};
    const int ar = (nt << 4) + lo;          // A: lane l -> row M = l%16
#pragma unroll
    for (int kk = 0; kk < NIN / 4; ++kk) {
      // A layout (16x4): lane l, vgpr v -> A[M=l%16][K = 4kk + v + 2*(l/16)]
      const v2f a = *(const v2f*)(&sAgg[ar * NIN + swz(ar, 4 * kk + 2 * hi)]);
      c = __builtin_amdgcn_wmma_f32_16x16x4_f32(
              false, a, false, b[kk], (short)0, c, false, false);
    }
    // C/D layout: lane l, vgpr v -> [M = v + 8*(l/16)][N = l%16]
#pragma unroll
    for (int v = 0; v < 8; ++v) {
      const int row = tileNode + v + (hi << 3);
      float val = c[v];
      if (ELU) val = (val > 0.0f) ? val : expm1f(val);   // jax.nn.elu
      dst[(size_t)row * NOUT + col] = val;
    }
  }
}

extern "C" void kernel_launch(void* const* d_in, const int* in_sizes, int n_in,
                              void* d_out, int out_size, void* d_ws, size_t ws_size,
                              hipStream_t stream) {
  // setup_inputs order: x, adj1_idx, adj1_w1, adj1_w2, adj2_idx, adj2_w1,
  // adj2_w2, W1, W2, r1_*..., r2_*...   (relation params are dead code: the
  // reference discards m1/m2, so only w2-weighted means + W1/W2 are live.)
  const float* x    = (const float*)d_in[0];
  const int*   idx1 = (const int*)  d_in[1];
  const float* w2a  = (const float*)d_in[3];
  const int*   idx2 = (const int*)  d_in[4];
  const float* w2b  = (const float*)d_in[6];
  const float* W1   = (const float*)d_in[7];
  const float* W2   = (const float*)d_in[8];
  float* out = (float*)d_out;
  float* h1  = (float*)d_ws;                 // N*128 f32 = 25.6 MB scratch

  constexpr int g1 = (N_NODES + 127) / 128;  // 391 blocks, 8 waves each
  constexpr int g2 = (N_NODES + 63) / 64;    // 782 blocks, 4 waves each

  lrgcn_stage<128, 8, NFEAT, NHID,  true ><<<g1, 256, 0, stream>>>(
      x, idx1, w2a, W1, h1);
  lrgcn_stage< 64, 4, NHID,  NCLASS,false><<<g2, 128, 0, stream>>>(
      h1, idx2, w2b, W2, out);
}